// DigitCaps_89610197664523
// MI455X (gfx1250) — compile-verified
//
#include <hip/hip_runtime.h>

#define Bn 100
#define Rn 1152
#define Cn 10
#define On 16
#define In 8
#define MP 112   // padded batch rows for WMMA m-tiles (7 * 16)

typedef __attribute__((ext_vector_type(2))) float v2f;
typedef __attribute__((ext_vector_type(4))) float v4f;
typedef __attribute__((ext_vector_type(8))) float v8f;

// -------------------------------------------------------------------------
// u_hat[b,r,c,o] = sum_i W[r,c,o,i] * x0[b,r,i]
// Per (r, mtile): D(16x16) += A(16x4) x B(4x16), two K-steps (I=8).
// A layout (ISA 32-bit A 16x4): lanes 0-15 -> {K=0,K=1}, lanes 16-31 -> {K=2,K=3}
// B layout (mirror): lanes0-15 rows K=0/1 @N=lane, lanes16-31 rows K=2/3
// D layout: VGPR v, lanes0-15 -> M=v, lanes16-31 -> M=v+8
// uhat is padded to MP=112 batch rows so ALL stores are unconditional
// (no EXEC-mask store chains); rows >= 100 are garbage and never read.
// -------------------------------------------------------------------------
__global__ __launch_bounds__(256) void uhat_wmma_kernel(
    const float* __restrict__ x0, const float* __restrict__ W,
    float* __restrict__ uhat)
{
  const int MT = MP / 16;  // 7 m-tiles
  const int gw = blockIdx.x * (blockDim.x >> 5) + (threadIdx.x >> 5);
  const int r  = gw / MT;
  const int mt = gw % MT;

  const int lane = threadIdx.x & 31;
  const int half = lane >> 4;
  const int sub  = lane & 15;

  int m = mt * 16 + sub;
  if (m > Bn - 1) m = Bn - 1;  // branch-free clamp for loads; EXEC stays full

  const float* abase = x0 + ((size_t)m * Rn + r) * In + half * 2;
  const v2f a0 = *reinterpret_cast<const v2f*>(abase);      // K = half*2 .. +1
  const v2f a1 = *reinterpret_cast<const v2f*>(abase + 4);  // K = 4+half*2 ..

  // store base for this wave: row mt*16 + half*8, column sub
  const int mrow0 = mt * 16 + half * 8;

  for (int c = 0; c < Cn; ++c) {
    const float* bbase = W + (((size_t)r * Cn + c) * On + sub) * In + half * 2;
    const v2f b0 = *reinterpret_cast<const v2f*>(bbase);
    const v2f b1 = *reinterpret_cast<const v2f*>(bbase + 4);
    v8f acc = {0.f, 0.f, 0.f, 0.f, 0.f, 0.f, 0.f, 0.f};
    acc = __builtin_amdgcn_wmma_f32_16x16x4_f32(false, a0, false, b0,
                                                (short)0, acc, false, false);
    acc = __builtin_amdgcn_wmma_f32_16x16x4_f32(false, a1, false, b1,
                                                (short)0, acc, false, false);
    float* dst = uhat + (((size_t)mrow0 * Rn + r) * Cn + c) * On + sub;
#pragma unroll
    for (int vv = 0; vv < 8; ++vv)
      dst[(size_t)vv * Rn * Cn * On] = acc[vv];   // unconditional, coalesced/lane
  }
}

// ----------------- softmax over R for each c (b_ij -> c_ij) -----------------
__global__ __launch_bounds__(256) void softmax_kernel(
    const float* __restrict__ bij, float* __restrict__ cij)
{
  const int c = blockIdx.x;
  const int t = threadIdx.x;
  __shared__ float red[256];

  float mx = -3.402823466e+38f;
  for (int r = t; r < Rn; r += 256) mx = fmaxf(mx, bij[r * Cn + c]);
  red[t] = mx; __syncthreads();
  for (int s = 128; s > 0; s >>= 1) {
    if (t < s) red[t] = fmaxf(red[t], red[t + s]);
    __syncthreads();
  }
  mx = red[0]; __syncthreads();

  float sum = 0.f;
  for (int r = t; r < Rn; r += 256) sum += __expf(bij[r * Cn + c] - mx);
  red[t] = sum; __syncthreads();
  for (int s = 128; s > 0; s >>= 1) {
    if (t < s) red[t] += red[t + s];
    __syncthreads();
  }
  const float inv = 1.0f / red[0];
  for (int r = t; r < Rn; r += 256)
    cij[r * Cn + c] = __expf(bij[r * Cn + c] - mx) * inv;
}

// ----------------- s_j[b,c,o] = sum_r c_ij[r,c]*u_hat[b,r,c,o] -------------
__global__ __launch_bounds__(256) void sj_kernel(
    const float* __restrict__ cij, const float* __restrict__ uhat,
    float* __restrict__ sj)
{
  const int bc = blockIdx.x;               // 0..Bn*Cn-1
  const int b = bc / Cn, c = bc % Cn;
  const int t = threadIdx.x;
  const int o = t & 15, rg = t >> 4;       // 16 r-groups
  float acc = 0.f;
  for (int r = rg; r < Rn; r += 16)
    acc += cij[r * Cn + c] * uhat[(((size_t)b * Rn + r) * Cn + c) * On + o];
  __shared__ float red[256];
  red[t] = acc; __syncthreads();
  for (int s = 128; s >= 16; s >>= 1) {
    if (t < s) red[t] += red[t + s];
    __syncthreads();
  }
  if (t < 16) sj[((size_t)b * Cn + c) * On + t] = red[t];
}

// ----------------- piecewise "squash": one thread per batch ----------------
__global__ void squash_kernel(const float* __restrict__ sj,
                              float* __restrict__ newf, float* __restrict__ v,
                              float* __restrict__ smut)
{
  const int b = blockIdx.x * blockDim.x + threadIdx.x;
  if (b >= Bn) return;
  const float T1 = -0.075410217f, T2 = 0.0f, T3 = 0.062207676f;

  float f[Cn];
  for (int c = 0; c < Cn; ++c) f[c] = sj[(b * Cn + c) * On + 0];

  // stable insertion sort (value, original index)
  float srt[Cn]; int ord[Cn];
  for (int p = 0; p < Cn; ++p) {
    float val = f[p]; int j = p;
    while (j > 0 && srt[j - 1] > val) { srt[j] = srt[j - 1]; ord[j] = ord[j - 1]; --j; }
    srt[j] = val; ord[j] = p;
  }
  int i1 = 0, i2 = 0, i3 = 0;
  for (int p = 0; p < Cn; ++p) {
    i1 += srt[p] < T1; i2 += srt[p] < T2; i3 += srt[p] < T3;
  }
  float nf[Cn];
  for (int p = 0; p < Cn; ++p) {
    const float s = srt[p];
    float ns = s;
    if (i1 > 0 && p < i1 - 1)                                ns = -0.074520095f * s + 0.349297946f;
    else if (i2 > 0 && i2 > i1 && p >= i1 && p < i2 - 1)     ns = -0.534473989f * s + 0.27196494f;
    else if (i3 > 0 && i3 > i2 && p >= i2 && p < i3 - 1)     ns =  0.637642944f * s + 0.295330779f;
    else if (i3 < Cn && p >= i3 && p < Cn - 1)               ns =  0.169344703f * s + 0.353784456f;
    nf[ord[p]] = ns;
  }
  for (int c = 0; c < Cn; ++c) {
    const float scale = nf[c];
    newf[b * Cn + c] = scale;
    for (int o = 0; o < On; ++o) {
      const float sm = (o == 0) ? scale : sj[(b * Cn + c) * On + o];
      smut[(b * Cn + c) * On + o] = sm;
      v[(b * Cn + c) * On + o] = scale * sm;
    }
  }
}

// ---- b_ij[r,c] += (1/B) * sum_b sum_o u_hat[b,r,c,o]*v[b,c,o] -------------
__global__ __launch_bounds__(160) void bupdate_kernel(
    const float* __restrict__ uhat, const float* __restrict__ v,
    float* __restrict__ bij)
{
  const int r = blockIdx.x;
  const int t = threadIdx.x;         // t = c*16 + o
  const int c = t >> 4, o = t & 15;
  float acc = 0.f;
  for (int b = 0; b < Bn; ++b)
    acc += uhat[((size_t)b * Rn + r) * (Cn * On) + t] * v[(b * Cn + c) * On + o];
  for (int m = 8; m >= 1; m >>= 1) acc += __shfl_xor(acc, m, 16);
  if (o == 0) bij[r * Cn + c] += acc * (1.0f / Bn);
}

// ----------------- output broadcast / copy kernels (NT streaming) ----------
__global__ void cfull_kernel(const float* __restrict__ cij, float* __restrict__ out)
{
  const int idx = blockIdx.x * blockDim.x + threadIdx.x;
  if (idx < Bn * Rn * Cn)
    __builtin_nontemporal_store(cij[idx % (Rn * Cn)], out + idx);
}

// Wb: W (5.9 MB, L2-resident) replicated Bn times: 590 MB of NT float4 streams.
__global__ void wb_kernel(const v4f* __restrict__ W4, v4f* __restrict__ out4)
{
  const int i = blockIdx.x * blockDim.x + threadIdx.x;   // 0..368639
  const size_t b = blockIdx.y;
  __builtin_nontemporal_store(W4[i], out4 + b * (size_t)(Rn * Cn * On * In / 4) + i);
}

// squashed_u[b,r,c,i] = x0[b,r,i]: per (b,r): 20 float4 out <- 2 float4 in.
__global__ void squashedu_kernel(const v4f* __restrict__ x04, v4f* __restrict__ out4)
{
  const int idx = blockIdx.x * blockDim.x + threadIdx.x;
  if (idx >= Bn * Rn * Cn * In / 4) return;
  const int hi = idx & 1;        // which half of the 8 floats (20 is even)
  const int br = idx / 20;       // (b*Rn + r)
  __builtin_nontemporal_store(x04[br * 2 + hi], out4 + idx);
}

__global__ void copy_vs_kernel(const float* __restrict__ v, const float* __restrict__ smut,
                               float* __restrict__ out0, float* __restrict__ out6)
{
  const int i = blockIdx.x * blockDim.x + threadIdx.x;
  if (i < Bn * Cn * On) { out0[i] = v[i]; out6[i] = smut[i]; }
}

// ---------------------------------------------------------------------------
extern "C" void kernel_launch(void* const* d_in, const int* in_sizes, int n_in,
                              void* d_out, int out_size, void* d_ws, size_t ws_size,
                              hipStream_t stream)
{
  (void)in_sizes; (void)n_in; (void)out_size; (void)ws_size;
  const float* x0 = (const float*)d_in[0];
  const float* x1 = (const float*)d_in[1];
  const float* x2 = (const float*)d_in[2];
  const float* W  = (const float*)d_in[3];
  float* out = (float*)d_out;
  float* ws  = (float*)d_ws;

  // workspace carve-out (floats): ~83 MB total (uhat padded to MP rows)
  float* uhat = ws;                                   // MP*Rn*Cn*On = 20,643,840
  float* bij  = uhat + (size_t)MP * Rn * Cn * On;     // Rn*Cn
  float* cij  = bij + Rn * Cn;                        // Rn*Cn
  float* sj   = cij + Rn * Cn;                        // Bn*Cn*On
  float* newf = sj + Bn * Cn * On;                    // Bn*Cn
  float* vj   = newf + Bn * Cn;                       // Bn*Cn*On
  float* smut = vj + Bn * Cn * On;                    // Bn*Cn*On

  // output slices (flat, in tuple order)
  float* out_v    = out;                 // v_j[:,0]      16,000
  float* out_cful = out + 16000;         // c_full        1,152,000
  float* out_Wb   = out + 1168000;       // Wb            147,456,000
  float* out_squ  = out + 148624000;     // squashed_u    9,216,000
  float* out_x1   = out + 157840000;     // x1            921,600
  float* out_x2   = out + 158761600;     // x2            78,400
  float* out_smut = out + 158840000;     // s_mut[:,0]    16,000

  // 1) u_hat via f32 WMMA: Rn*7 = 8064 waves, 8 waves/block
  uhat_wmma_kernel<<<Rn * (MP / 16) / 8, 256, 0, stream>>>(x0, W, uhat);

  // 2) b_ij = 0 each call (deterministic; graph-capture safe)
  hipMemsetAsync(bij, 0, Rn * Cn * sizeof(float), stream);

  // 3) three routing iterations
  for (int it = 0; it < 3; ++it) {
    softmax_kernel<<<Cn, 256, 0, stream>>>(bij, cij);
    sj_kernel<<<Bn * Cn, 256, 0, stream>>>(cij, uhat, sj);
    squash_kernel<<<1, 128, 0, stream>>>(sj, newf, vj, smut);
    if (it < 2) bupdate_kernel<<<Rn, 160, 0, stream>>>(uhat, vj, bij);
  }

  // 4) outputs (memory-bound: ~635 MB of NT stores at HBM rate)
  copy_vs_kernel<<<(Bn * Cn * On + 255) / 256, 256, 0, stream>>>(vj, smut, out_v, out_smut);
  cfull_kernel<<<(Bn * Rn * Cn + 255) / 256, 256, 0, stream>>>(cij, out_cful);
  wb_kernel<<<dim3(Rn * Cn * On * In / 4 / 256, Bn), 256, 0, stream>>>(
      (const v4f*)W, (v4f*)out_Wb);
  squashedu_kernel<<<(Bn * Rn * Cn * In / 4 + 255) / 256, 256, 0, stream>>>(
      (const v4f*)x0, (v4f*)out_squ);
  hipMemcpyAsync(out_x1, x1, (size_t)Bn * Rn * In * sizeof(float),
                 hipMemcpyDeviceToDevice, stream);
  hipMemcpyAsync(out_x2, x2, (size_t)Bn * 28 * 28 * sizeof(float),
                 hipMemcpyDeviceToDevice, stream);
}